// GraphConvolution_9259949490534
// MI455X (gfx1250) — compile-verified
//
#include <hip/hip_runtime.h>

typedef __attribute__((ext_vector_type(2))) float v2f;
typedef __attribute__((ext_vector_type(8))) float v8f;

#define GCN_N 100000
#define GCN_E 3200000
#define GCN_F 128
#define LT 132   // transposed-W LDS row stride (dwords): bank step 4 =>
                 // ds_load_b64 from both 16-lane halves covers all 64 banks

// ---------------------------------------------------------------------------
// Kernel 1: support = X @ W via V_WMMA_F32_16X16X4_F32 (fp32 WMMA, wave32).
// Block = 256 thr (8 waves). W is staged TRANSPOSED into LDS (Wt[n][k],
// 67.6KB; CDNA5 WGP has 320KB LDS) so each lane's B pair {W[k][n],W[k+1][n]}
// is one aligned ds_load_b64 straight into the WMMA operand pair.
// Each wave owns a 16-row strip of X and computes all 8 N-tiles:
// one A load feeds 8 WMMAs.
// ---------------------------------------------------------------------------
__global__ __launch_bounds__(256) void gcn_gemm_wmma(
    const float* __restrict__ X, const float* __restrict__ W,
    float* __restrict__ S) {
  __shared__ float ldsWt[GCN_F * LT];   // Wt[n][k] = W[k][n]

  const int tid  = threadIdx.x;
  const int wave = tid >> 5;
  const int lane = tid & 31;
  const int l16  = lane & 15;
  const int half = lane >> 4;

  // ---- stage W[k][n] -> ldsWt[n*LT + k] (coalesced global read) ----
#pragma unroll
  for (int idx = tid * 4; idx < GCN_F * GCN_F; idx += 256 * 4) {
    const float4 v = *(const float4*)(W + idx);
    const int k = idx >> 7;        // row of W
    const int n = idx & 127;       // col of W (multiple of 4)
    ldsWt[(n + 0) * LT + k] = v.x;
    ldsWt[(n + 1) * LT + k] = v.y;
    ldsWt[(n + 2) * LT + k] = v.z;
    ldsWt[(n + 3) * LT + k] = v.w;
  }
  __syncthreads();

  const int m0 = (blockIdx.x * 8 + wave) << 4;   // 16-row strip per wave
  if (m0 < GCN_N) {
    // A layout (16x4 f32): lanes 0-15 -> K={0,1}, lanes 16-31 -> K={2,3}
    const float* arow = X + (size_t)(m0 + l16) * GCN_F + 2 * half;

    v8f acc[8];
#pragma unroll
    for (int n = 0; n < 8; ++n) acc[n] = (v8f){};

#pragma unroll 2
    for (int k0 = 0; k0 < GCN_F; k0 += 4) {
      const v2f a = *(const v2f*)(arow + k0);    // 8B contiguous global load
#pragma unroll
      for (int n = 0; n < 8; ++n) {
        // B 4x16: lane%16 = column n*16+l16; half selects K pair {0,1}/{2,3}
        const v2f b =
            *(const v2f*)(&ldsWt[((n << 4) + l16) * LT + k0 + 2 * half]);
        acc[n] = __builtin_amdgcn_wmma_f32_16x16x4_f32(
            false, a, false, b, (short)0, acc[n], false, false);
      }
    }

    // C/D layout: VGPR r -> row m0 + r + 8*half, col n*16 + lane%16
    float* srow = S + (size_t)(m0 + 8 * half) * GCN_F + l16;
#pragma unroll
    for (int n = 0; n < 8; ++n)
#pragma unroll
      for (int r = 0; r < 8; ++r)
        srow[(size_t)r * GCN_F + (n << 4)] = acc[n][r];
  }
}

// ---------------------------------------------------------------------------
// Counting sort of edges by destination: histogram -> scan -> scatter
// ---------------------------------------------------------------------------
__global__ __launch_bounds__(256) void gcn_hist(
    const int* __restrict__ edst, unsigned* __restrict__ counts) {
  int e = blockIdx.x * blockDim.x + threadIdx.x;
  if (e < GCN_E) atomicAdd(&counts[edst[e]], 1u);
}

__global__ __launch_bounds__(1024) void gcn_scan(
    const unsigned* __restrict__ counts, unsigned* __restrict__ offs,
    unsigned* __restrict__ cursor) {
  __shared__ unsigned lds[1024];
  __shared__ unsigned carry;
  if (threadIdx.x == 0) carry = 0;
  __syncthreads();
  for (int base = 0; base < GCN_N; base += 1024) {
    int i = base + (int)threadIdx.x;
    unsigned v = (i < GCN_N) ? counts[i] : 0u;
    lds[threadIdx.x] = v;
    __syncthreads();
    for (int off = 1; off < 1024; off <<= 1) {
      unsigned t = (threadIdx.x >= (unsigned)off) ? lds[threadIdx.x - off] : 0u;
      __syncthreads();
      lds[threadIdx.x] += t;
      __syncthreads();
    }
    unsigned excl = lds[threadIdx.x] - v;
    if (i < GCN_N) {
      unsigned o = carry + excl;
      offs[i] = o;
      cursor[i] = o;
    }
    __syncthreads();
    if (threadIdx.x == 1023) carry += lds[1023];
    __syncthreads();
  }
  if (threadIdx.x == 0) offs[GCN_N] = carry;
}

__global__ __launch_bounds__(256) void gcn_scatter(
    const int* __restrict__ esrc, const int* __restrict__ edst,
    const float* __restrict__ ew, unsigned* __restrict__ cursor,
    int* __restrict__ ssrc, float* __restrict__ sw) {
  int e = blockIdx.x * blockDim.x + threadIdx.x;
  if (e < GCN_E) {
    int d = edst[e];
    unsigned p = atomicAdd(&cursor[d], 1u);
    ssrc[p] = esrc[e];
    sw[p] = ew[e];
  }
}

// ---------------------------------------------------------------------------
// One wave per destination node. float4/lane = 128 features per wave.
// Support rows live in the 192MB L2; global_prefetch keeps ~4 row fetches
// in flight ahead of the FMA stream to hide L2 latency.
// ---------------------------------------------------------------------------
__global__ __launch_bounds__(256) void gcn_aggregate(
    const float* __restrict__ S, const unsigned* __restrict__ offs,
    const int* __restrict__ ssrc, const float* __restrict__ sw,
    const float* __restrict__ bias, float* __restrict__ out) {
  int node = blockIdx.x * (blockDim.x >> 5) + (threadIdx.x >> 5);
  int lane = threadIdx.x & 31;
  if (node >= GCN_N) return;

  unsigned beg = offs[node];
  unsigned end = offs[node + 1];

  float4 acc = make_float4(0.f, 0.f, 0.f, 0.f);
  for (unsigned i = beg; i < end; ++i) {
    if (i + 4 < end) {  // prefetch support row 4 edges ahead (L2 resident)
      int sp = ssrc[i + 4];
      __builtin_prefetch(S + (size_t)sp * GCN_F + lane * 4, 0, 3);
    }
    int s = ssrc[i];       // wave-uniform (hardware broadcasts the load)
    float wgt = sw[i];
    const float4 v = *(const float4*)(S + (size_t)s * GCN_F + lane * 4);
    acc.x += wgt * v.x;
    acc.y += wgt * v.y;
    acc.z += wgt * v.z;
    acc.w += wgt * v.w;
  }
  const float4 b = *(const float4*)(bias + lane * 4);
  acc.x += b.x; acc.y += b.y; acc.z += b.z; acc.w += b.w;
  *(float4*)(out + (size_t)node * GCN_F + lane * 4) = acc;
}

// ---------------------------------------------------------------------------
// Fallback path (small workspace): out = bias, then per-edge f32 atomics.
// ---------------------------------------------------------------------------
__global__ __launch_bounds__(256) void gcn_init_bias(
    const float* __restrict__ bias, float* __restrict__ out) {
  int i = blockIdx.x * blockDim.x + threadIdx.x;
  if (i < GCN_N * GCN_F) out[i] = bias[i & (GCN_F - 1)];
}

__global__ __launch_bounds__(256) void gcn_edge_atomic(
    const float* __restrict__ S, const int* __restrict__ esrc,
    const int* __restrict__ edst, const float* __restrict__ ew,
    float* __restrict__ out) {
  int e = blockIdx.x * (blockDim.x >> 5) + (threadIdx.x >> 5);
  int lane = threadIdx.x & 31;
  if (e >= GCN_E) return;
  int s = esrc[e];
  int d = edst[e];
  float wgt = ew[e];
  const float4 v = *(const float4*)(S + (size_t)s * GCN_F + lane * 4);
  float* o = out + (size_t)d * GCN_F + lane * 4;
  unsafeAtomicAdd(o + 0, wgt * v.x);
  unsafeAtomicAdd(o + 1, wgt * v.y);
  unsafeAtomicAdd(o + 2, wgt * v.z);
  unsafeAtomicAdd(o + 3, wgt * v.w);
}

// ---------------------------------------------------------------------------
extern "C" void kernel_launch(void* const* d_in, const int* in_sizes, int n_in,
                              void* d_out, int out_size, void* d_ws, size_t ws_size,
                              hipStream_t stream) {
  const float* X    = (const float*)d_in[0];  // [N, 128]
  const float* W    = (const float*)d_in[1];  // [128, 128]
  const float* bias = (const float*)d_in[2];  // [128]
  const int*   esrc = (const int*)d_in[3];    // [E]
  const int*   edst = (const int*)d_in[4];    // [E]
  const float* ew   = (const float*)d_in[5];  // [E]
  float* out = (float*)d_out;

  auto align256 = [](size_t x) { return (x + 255) & ~(size_t)255; };
  char* ws = (char*)d_ws;
  size_t supB  = align256((size_t)GCN_N * GCN_F * sizeof(float)); // 51.2 MB
  size_t ssrcB = align256((size_t)GCN_E * sizeof(int));           // 12.8 MB
  size_t swB   = align256((size_t)GCN_E * sizeof(float));         // 12.8 MB
  size_t cntB  = align256((size_t)GCN_N * sizeof(unsigned));
  size_t offB  = align256((size_t)(GCN_N + 1) * sizeof(unsigned));
  size_t needSorted = supB + ssrcB + swB + cntB + offB +
                      align256((size_t)GCN_N * sizeof(unsigned));  // ~78.1 MB

  float* S = (float*)ws;

  // ---- Phase 1: dense GEMM via fp32 WMMA (transposed-LDS W, A reuse x8) ----
  gcn_gemm_wmma<<<(GCN_N / 16 + 7) / 8, 256, 0, stream>>>(X, W, S);

  if (ws_size >= needSorted) {
    int*      ssrc   = (int*)(ws + supB);
    float*    sw     = (float*)(ws + supB + ssrcB);
    unsigned* counts = (unsigned*)(ws + supB + ssrcB + swB);
    unsigned* offs   = (unsigned*)(ws + supB + ssrcB + swB + cntB);
    unsigned* cursor = (unsigned*)(ws + supB + ssrcB + swB + cntB + offB);

    hipMemsetAsync(counts, 0, (size_t)GCN_N * sizeof(unsigned), stream);
    gcn_hist<<<(GCN_E + 255) / 256, 256, 0, stream>>>(edst, counts);
    gcn_scan<<<1, 1024, 0, stream>>>(counts, offs, cursor);
    gcn_scatter<<<(GCN_E + 255) / 256, 256, 0, stream>>>(esrc, edst, ew,
                                                         cursor, ssrc, sw);
    gcn_aggregate<<<(GCN_N + 7) / 8, 256, 0, stream>>>(S, offs, ssrc, sw,
                                                       bias, out);
  } else {
    gcn_init_bias<<<(GCN_N * GCN_F + 255) / 256, 256, 0, stream>>>(bias, out);
    gcn_edge_atomic<<<(GCN_E + 7) / 8, 256, 0, stream>>>(S, esrc, edst, ew, out);
  }
}